// M_Z_40415642255478
// MI455X (gfx1250) — compile-verified
//
#include <hip/hip_runtime.h>

// Problem dims
#define B_  2
#define P_  128
#define N_  2048
#define M_  64     // == I (modes) == mode-mix size
#define T_  3

#define WAVES   16
#define THREADS (WAVES * 32)
#define NITER   4               // n-values per workgroup (amortize sA + weight L2 traffic)

// LDS strides (floats), padded for bank-conflict-free WMMA fragment access
#define SA_STRIDE 132           // DFT matrix: 128 rows(i') x 128 cols(p)
#define SB_STRIDE 200           // f slice:   128 rows(p)  x 192 cols(l*3+t)
#define SG_STRIDE 194           // G/G':      128 rows(i') x 192 cols(l*3+t)
#define SB_OFF (128 * SA_STRIDE)
#define SG_OFF (SB_OFF + 128 * SB_STRIDE)
#define ST_OFF (SG_OFF + 128 * SG_STRIDE)
#define SMEM_FLOATS (ST_OFF + WAVES * 384)
#define SMEM_BYTES  (SMEM_FLOATS * 4)     // 293,888 B <= 320 KB

typedef float v2f __attribute__((ext_vector_type(2)));
typedef float v8f __attribute__((ext_vector_type(8)));

static __device__ __forceinline__ v8f wmma_f32_16x16x4(v2f a, v2f b, v8f c) {
  // 8 args: (neg_a, A, neg_b, B, c_mod, C, reuse_a, reuse_b)
  return __builtin_amdgcn_wmma_f32_16x16x4_f32(false, a, false, b, (short)0, c,
                                               false, false);
}

__global__ void __launch_bounds__(THREADS)
fno_fused_kernel(const float* __restrict__ fz,   // (B,P,N,M,T) f32
                 const float* __restrict__ wt,   // (I,M,M) complex64 -> float pairs
                 float* __restrict__ out) {      // (B,N,I,M,T) complex64 -> float pairs
  extern __shared__ float smem[];
  float* sA = smem;            // DFT matrix, rows 0..63 = Re, 64..127 = Im
  float* sB = smem + SB_OFF;   // f slice for current n
  float* sG = smem + SG_OFF;   // truncated-FFT result, then in-place ifft3
  float* sS = smem + ST_OFF;   // per-wave output staging (384 floats each)

  const int tid  = threadIdx.x;
  const int lane = tid & 31;
  const int wave = tid >> 5;
  const int bq   = blockIdx.y;

  // ---- build truncated DFT matrix: F[i] = sum_p f[p] * e^{-2*pi*j*i*p/128} ----
  for (int e = tid; e < 64 * 128; e += THREADS) {
    int i = e >> 7, p = e & 127;
    int m = (i * p) & 127;                       // exact twiddle-index reduction
    float ang = -6.28318530717958647692f * (float)m * (1.0f / 128.0f);
    float s, c;
    __sincosf(ang, &s, &c);
    sA[i * SA_STRIDE + p]        = c;            // Re row
    sA[(i + 64) * SA_STRIDE + p] = s;            // Im row (sin of negative angle)
  }

  for (int it = 0; it < NITER; ++it) {
    const int n = blockIdx.x * NITER + it;
    __syncthreads();   // previous iteration finished reading sB/sG (also orders sA fill)

    // ---- load f[b, p=0..127, n, l, t] -> sB[p][l*3+t], coalesced float4 ----
    const float* fb = fz + ((size_t)bq * P_ * N_ + (size_t)n) * (M_ * T_);
    for (int r = 0; r < 12; ++r) {               // 128*48 float4 / 512 threads
      int lin = r * THREADS + tid;
      int p = lin / 48, q = (lin % 48) * 4;
      *(float4*)(sB + p * SB_STRIDE + q) =
          *(const float4*)(fb + (size_t)p * (N_ * M_ * T_) + q);
    }
    if (it + 1 < NITER) {                        // hint next n's slice into cache
      int p = tid & 127, off = (tid >> 7) * 48;
      __builtin_prefetch(fb + (size_t)p * (N_ * M_ * T_) + (M_ * T_) + off, 0, 1);
    }
    __syncthreads();

    // ---- stage 1: G[i',col] = sA[i',p] @ sB[p,col] ----
    // wave -> 1 row-tile x 6 col-tiles: A-fragment loaded once per k-step,
    // amortized over 6 WMMAs (6 live accumulators).
    {
      const int rt = wave >> 1;                  // 8 row tiles, 2 waves each
      const int ch = (wave & 1) * 6;             // first of this wave's 6 col tiles
      const int kofs = 2 * (lane >> 4);
      const float* Ar = sA + (rt * 16 + (lane & 15)) * SA_STRIDE + kofs;
      const float* Bc = sB + ch * 16 + (lane & 15) + kofs * SB_STRIDE;
      v8f acc[6] = {};
#pragma unroll 2
      for (int ks = 0; ks < 32; ++ks) {
        v2f a = { Ar[4 * ks], Ar[4 * ks + 1] };
#pragma unroll
        for (int c = 0; c < 6; ++c) {
          v2f b = { Bc[(4 * ks) * SB_STRIDE + c * 16],
                    Bc[(4 * ks + 1) * SB_STRIDE + c * 16] };
          acc[c] = wmma_f32_16x16x4(a, b, acc[c]);
        }
      }
      float* g = sG + (rt * 16 + (lane >> 4) * 8) * SG_STRIDE + ch * 16 + (lane & 15);
#pragma unroll
      for (int c = 0; c < 6; ++c)
#pragma unroll
        for (int v = 0; v < 8; ++v) g[v * SG_STRIDE + c * 16] = acc[c][v];
    }
    __syncthreads();

    // ---- fold the 3-point ifft (over t) into G, in place ----
    for (int e = tid; e < 64 * 64; e += THREADS) {
      int i = e >> 6, l = e & 63;
      float* gr = sG + i * SG_STRIDE + l * 3;
      float* gi = sG + (i + 64) * SG_STRIDE + l * 3;
      float r0 = gr[0], r1 = gr[1], r2 = gr[2];
      float q0 = gi[0], q1 = gi[1], q2 = gi[2];
      const float C = -0.5f, S = 0.86602540378443864676f, K3 = (1.0f / 3.0f);
      gr[0] = K3 * (r0 + r1 + r2);
      gi[0] = K3 * (q0 + q1 + q2);
      gr[1] = K3 * (r0 + C * r1 - S * q1 + C * r2 + S * q2);
      gi[1] = K3 * (q0 + C * q1 + S * r1 + C * q2 - S * r2);
      gr[2] = K3 * (r0 + C * r1 + S * q1 + C * r2 - S * q2);
      gi[2] = K3 * (q0 + C * q1 - S * r1 + C * q2 + S * r2);
    }
    __syncthreads();

    // ---- stage 2: per mode i, complex 64x64 mix as real GEMM ----
    // rows = (t',re/im)=6 (padded to 16), cols = s, K = (l,re/im)=128.
    // K-halves k and k+64 share one float2 weight load and two sG reads.
    const int row  = lane & 15;
    const int prt  = (row >= 3) ? 1 : 0;         // 0 = D_re rows, 1 = D_im rows
    const int tpr  = row - prt * 3;
    const int tpc  = (tpr > 2) ? 0 : tpr;        // clamp for padded rows (masked to 0)
    const int valid = (row < 6);
    const int kofs = 2 * (lane >> 4);
    for (int mi = 0; mi < 4; ++mi) {
      const int i = wave * 4 + mi;               // 64 modes / 16 waves
      float* stg = sS + wave * 384;
      const float* gR = sG + i * SG_STRIDE + tpc;          // G're row base
      const float* gI = sG + (i + 64) * SG_STRIDE + tpc;   // G'im row base
      for (int ct = 0; ct < 4; ++ct) {
        const int s = ct * 16 + (lane & 15);
        const float* wb = wt + (size_t)(i * 64 + s) * 128; // weights[i][s][:], interleaved
        v8f acc = {};
#pragma unroll 4
        for (int ks = 0; ks < 16; ++ks) {        // covers k and k+64 together
          v2f a0, a1, b0, b1;
#pragma unroll
          for (int v = 0; v < 2; ++v) {
            int l = 4 * ks + kofs + v;           // 0..63
            float2 w2 = *(const float2*)(wb + 2 * l);  // {Re, Im}, 8B load
            float gr = gR[l * 3];
            float gi = gI[l * 3];
            // A2 low half (kp=0): re-rows <- G'r, im-rows <- G'i
            float av0 = prt ? gi : gr;
            // A2 high half (kp=1): re-rows <- -G'i, im-rows <- G'r
            float av1 = prt ? gr : -gi;
            a0[v] = valid ? av0 : 0.0f;
            a1[v] = valid ? av1 : 0.0f;
            b0[v] = w2.x;                        // B2 low half: Wr^T
            b1[v] = w2.y;                        // B2 high half: Wi^T
          }
          acc = wmma_f32_16x16x4(a0, b0, acc);
          acc = wmma_f32_16x16x4(a1, b1, acc);
        }
        if (lane < 16) {                         // lanes 16-31 hold padded rows 8-15
#pragma unroll
          for (int v = 0; v < 6; ++v) {
            int pp = (v >= 3) ? 1 : 0;
            int tt = v - pp * 3;
            stg[(ct * 16 + lane) * 6 + tt * 2 + pp] = acc[v];
          }
        }
      }
      // coalesced store: 64 s * 3 t' * 2 parts = 384 floats = 96 float4
      float4* ov = (float4*)(out + ((size_t)((bq * N_ + n) * 64 + i)) * 384);
      const float4* sv = (const float4*)stg;
#pragma unroll
      for (int r = 0; r < 3; ++r) ov[r * 32 + lane] = sv[r * 32 + lane];
    }
  }
}

extern "C" void kernel_launch(void* const* d_in, const int* in_sizes, int n_in,
                              void* d_out, int out_size, void* d_ws, size_t ws_size,
                              hipStream_t stream) {
  (void)in_sizes; (void)n_in; (void)out_size; (void)d_ws; (void)ws_size;
  const float* fz = (const float*)d_in[0];
  const float* wt = (const float*)d_in[1];   // complex64 viewed as float pairs
  float* out = (float*)d_out;

  static_assert(SMEM_BYTES <= 320 * 1024, "LDS budget exceeded");
  (void)hipFuncSetAttribute(reinterpret_cast<const void*>(fno_fused_kernel),
                            hipFuncAttributeMaxDynamicSharedMemorySize, SMEM_BYTES);
  dim3 grid(N_ / NITER, B_);
  fno_fused_kernel<<<grid, dim3(THREADS), SMEM_BYTES, stream>>>(fz, wt, out);
}